// ScatteringTransform_45878840656425
// MI455X (gfx1250) — compile-verified
//
#include <hip/hip_runtime.h>

// Scattering transform for MI455X (gfx1250, wave32, WMMA).
// B=16, F=16, N=1024, J=8, L=3. Output phi: (16,16,73) f32.
//
// 38.7 GFLOP of f32 GEMM vs ~50 MB mandatory traffic -> compute-bound on the
// f32 matrix pipe (V_WMMA_F32_16X16X4_F32, matches reference precision).
// Layer-2 result S2 (64 MB) feeds only the lowpass projection, which is fused
// into the GEMM epilogue, so S2 never touches memory.
// psi is pre-transposed once (ws scratch) so B fragments load as contiguous
// b64 per lane; each wave computes 4 n-tiles per A-fragment load (4 indep
// WMMA chains, ~1.25 vmem per WMMA).

typedef __attribute__((ext_vector_type(2))) float v2f;
typedef __attribute__((ext_vector_type(8))) float v8f;

#define N_DIM 1024
#define B_DIM 16
#define F_DIM 16
#define J_DIM 8
#define C_OUT 73   // 1 + 8 + 64

__device__ __forceinline__ v8f wmma_f32(v2f a, v2f b, v8f c) {
  // 8 args: (neg_a, A, neg_b, B, c_mod, C, reuse_a, reuse_b)
  return __builtin_amdgcn_wmma_f32_16x16x4_f32(
      false, a, false, b, (short)0, c, false, false);
}

// phi0[b,f] = dot(x[b,f,:], lowpass); one wave per (b,f) row.
__global__ void __launch_bounds__(256)
phi0_kernel(const float* __restrict__ x, const float* __restrict__ lp,
            float* __restrict__ out) {
  int wave = (blockIdx.x * blockDim.x + threadIdx.x) >> 5;
  int lane = threadIdx.x & 31;
  if (wave >= B_DIM * F_DIM) return;
  const float* __restrict__ px = x + (size_t)wave * N_DIM;
  float s = 0.f;
  #pragma unroll 4
  for (int i = lane; i < N_DIM; i += 32) s += px[i] * lp[i];
  #pragma unroll
  for (int off = 16; off >= 1; off >>= 1) s += __shfl_xor(s, off, 32);
  if (lane == 0) out[(size_t)wave * C_OUT] = s;  // wave == b*F + f, channel 0
}

// Tiled transpose: psiT[j][n][k] = psi[j][k][n] (8 matrices of 1024x1024).
__global__ void __launch_bounds__(256)
transpose_psi(const float* __restrict__ psi, float* __restrict__ psiT) {
  __shared__ float tile[32][33];
  const int tx = threadIdx.x & 31;
  const int ty = threadIdx.x >> 5;                 // 8 row-groups of 32
  const float* __restrict__ src = psi + ((size_t)blockIdx.z << 20);
  float* __restrict__ dst = psiT + ((size_t)blockIdx.z << 20);
  const int x0 = blockIdx.x << 5, y0 = blockIdx.y << 5;
  #pragma unroll
  for (int r = 0; r < 32; r += 8)
    tile[ty + r][tx] = src[(size_t)(y0 + ty + r) * N_DIM + x0 + tx];
  __syncthreads();
  #pragma unroll
  for (int r = 0; r < 32; r += 8)
    dst[(size_t)(x0 + ty + r) * N_DIM + y0 + tx] = tile[tx][ty + r];
}

// One scattering layer: for each j, C = |A x psi_j| (A: Mrows x 1024).
// Wave = one (16-row mtile, j); loops 16 groups of 4 n-tiles.
// Per 4-k chunk: one A b64 per lane, reused by 4 B fragments / 4 WMMA chains.
// TRB=1: Bmat is psiT (b64 B loads). TRB=0: Bmat is psi (2x b32, fallback).
template <int KD, int CBASE, int DOSTORE, int TRB>
__global__ void __launch_bounds__(256)
scat_layer(const float* __restrict__ A, const float* __restrict__ Bmat,
           const float* __restrict__ lp, float* __restrict__ S_out,
           float* __restrict__ phi_out) {
  const int lane  = threadIdx.x & 31;
  const int j     = threadIdx.x >> 5;       // wave id == psi index
  const int row0  = blockIdx.x << 4;        // 16 A-rows per wave
  const int mrow  = lane & 15;
  const int khalf = (lane >> 4) << 1;       // 0 or 2 (fragment K split)

  const float* __restrict__ pA = A + (size_t)(row0 + mrow) * N_DIM + khalf;
  const float* __restrict__ Bj = Bmat + ((size_t)j << 20);

  float phirow[8];
  #pragma unroll
  for (int v = 0; v < 8; ++v) phirow[v] = 0.f;

  for (int n0 = 0; n0 < N_DIM; n0 += 64) {     // 4 n-tiles per iteration
    v8f acc[4] = {};
    const float* __restrict__ pBT =
        Bj + (size_t)(n0 + mrow) * N_DIM + khalf;          // TRB=1 base
    const float* __restrict__ pBd =
        Bj + (size_t)khalf * N_DIM + n0 + mrow;            // TRB=0 base

    #pragma unroll 4
    for (int k0 = 0; k0 < N_DIM; k0 += 8) {
      // A fragment: lane holds row M=lane&15, K = k0 + khalf + {0,1}
      v2f a0 = *(const v2f*)(pA + k0);
      v2f a1 = *(const v2f*)(pA + k0 + 4);
      v2f b0[4], b1[4];
      #pragma unroll
      for (int t = 0; t < 4; ++t) {
        if (TRB) {
          // psiT[n][k]: contiguous K pair -> single b64 per fragment
          b0[t] = *(const v2f*)(pBT + (size_t)t * 16 * N_DIM + k0);
          b1[t] = *(const v2f*)(pBT + (size_t)t * 16 * N_DIM + k0 + 4);
        } else {
          b0[t].x = pBd[(size_t)(k0    ) * N_DIM + t * 16];
          b0[t].y = pBd[(size_t)(k0 + 1) * N_DIM + t * 16];
          b1[t].x = pBd[(size_t)(k0 + 4) * N_DIM + t * 16];
          b1[t].y = pBd[(size_t)(k0 + 5) * N_DIM + t * 16];
        }
      }
      // 4 independent accumulation chains; dependent WMMAs are 4 apart.
      #pragma unroll
      for (int t = 0; t < 4; ++t) acc[t] = wmma_f32(a0, b0[t], acc[t]);
      #pragma unroll
      for (int t = 0; t < 4; ++t) acc[t] = wmma_f32(a1, b1[t], acc[t]);
    }

    // Epilogue: abs, fused lowpass projection, optional S1 store.
    #pragma unroll
    for (int t = 0; t < 4; ++t) {
      float lpv = lp[n0 + 16 * t + mrow];
      #pragma unroll
      for (int v = 0; v < 8; ++v) {
        // C/D layout: vgpr v, lane l -> M = v + 8*(l>>4), N = l&15
        float d = __builtin_fabsf(acc[t][v]);
        phirow[v] += d * lpv;
        if (DOSTORE) {
          int r = row0 + v + ((lane >> 4) << 3);               // A row b*16+f
          int srow = (((r >> 4) * J_DIM + j) << 4) + (r & 15); // (b*8+j)*16+f
          S_out[(size_t)srow * N_DIM + n0 + 16 * t + mrow] = d;
        }
      }
    }
  }

  // Reduce each output row's lowpass partial across its 16 lanes.
  #pragma unroll
  for (int v = 0; v < 8; ++v) {
    float s = phirow[v];
    #pragma unroll
    for (int off = 1; off < 16; off <<= 1) s += __shfl_xor(s, off, 16);
    phirow[v] = s;
  }
  if ((lane & 15) == 0) {
    #pragma unroll
    for (int v = 0; v < 8; ++v) {
      int r = row0 + v + ((lane >> 4) << 3);
      int b = r / (KD * F_DIM);
      int k = (r / F_DIM) % KD;
      int f = r % F_DIM;
      int c = CBASE + k * J_DIM + j;
      phi_out[(size_t)(b * F_DIM + f) * C_OUT + c] = phirow[v];
    }
  }
}

extern "C" void kernel_launch(void* const* d_in, const int* in_sizes, int n_in,
                              void* d_out, int out_size, void* d_ws, size_t ws_size,
                              hipStream_t stream) {
  const float* x   = (const float*)d_in[0];   // (16,16,1024)
  const float* psi = (const float*)d_in[1];   // (8,1024,1024)
  const float* lp  = (const float*)d_in[2];   // (1024,)
  float* out = (float*)d_out;                 // (16,16,73)

  const size_t s1_bytes  = (size_t)B_DIM * J_DIM * F_DIM * N_DIM * 4; // 8 MB
  const size_t psiT_bytes = (size_t)J_DIM * N_DIM * N_DIM * 4;        // 32 MB
  float* S1   = (float*)d_ws;
  float* psiT = (float*)((char*)d_ws + s1_bytes);
  const bool useT = ws_size >= s1_bytes + psiT_bytes;

  // channel 0
  phi0_kernel<<<B_DIM * F_DIM / 8, 256, 0, stream>>>(x, lp, out);

  if (useT) {
    transpose_psi<<<dim3(32, 32, 8), 256, 0, stream>>>(psi, psiT);
    // layer 1: A = x (256 rows), store S1, channels 1..8
    scat_layer<1, 1, 1, 1><<<(B_DIM * F_DIM) / 16, 256, 0, stream>>>(
        x, psiT, lp, S1, out);
    // layer 2: A = S1 (2048 rows), fused projection only, channels 9..72
    scat_layer<8, 9, 0, 1><<<(B_DIM * J_DIM * F_DIM) / 16, 256, 0, stream>>>(
        S1, psiT, lp, nullptr, out);
  } else {
    scat_layer<1, 1, 1, 0><<<(B_DIM * F_DIM) / 16, 256, 0, stream>>>(
        x, psi, lp, S1, out);
    scat_layer<8, 9, 0, 0><<<(B_DIM * J_DIM * F_DIM) / 16, 256, 0, stream>>>(
        S1, psi, lp, nullptr, out);
  }
}